// YOLOLoss_41755672051846
// MI455X (gfx1250) — compile-verified
//
#include <hip/hip_runtime.h>
#include <hip/hip_bf16.h>
#include <stdint.h>

// ---------------------------------------------------------------------------
// YOLO loss on MI455X (gfx1250): bandwidth-bound streaming reduction.
// Data path: TDM tensor_load_to_lds (async, TENSORcnt) double-buffered tiles.
// ---------------------------------------------------------------------------

#define FEAT            13
#define CELLS_PER_TILE  256
#define FLOATS_PER_TILE (CELLS_PER_TILE * FEAT)   // 3328
#define BYTES_PER_TILE  (FLOATS_PER_TILE * 4)     // 13312
#define TILES_PER_BLOCK 8
#define TOTAL_CELLS     (32768 * 7 * 7)           // 1,605,632
#define NTILES          (TOTAL_CELLS / CELLS_PER_TILE)   // 6272
#define NBLOCKS         (NTILES / TILES_PER_BLOCK)       // 784

typedef unsigned int u32x4 __attribute__((ext_vector_type(4)));
typedef int          i32x4 __attribute__((ext_vector_type(4)));
typedef int          i32x8 __attribute__((ext_vector_type(8)));

// Issue one TDM load of FLOATS_PER_TILE contiguous f32 from global into LDS.
// D# per CDNA5 ISA ch. 8: group0 = {flags, lds_addr, global_addr, type=2},
// group1 = {data_size=4B, tensor_dim0=n, tensor_dim1=1, tile_dim0=n,
//           tile_dim1=1, dim0_stride=n}. Groups 2/3 zero (2D tensor).
__device__ __forceinline__ void tdm_load_tile(const float* gsrc,
                                              unsigned lds_byte_off) {
  unsigned long long ga = (unsigned long long)(uintptr_t)gsrc;
  const unsigned n = FLOATS_PER_TILE;  // 3328, fits 16-bit tile_dim0

  u32x4 g0;
  g0.x = 1u;                                             // count=1 (valid), user mode
  g0.y = lds_byte_off;                                   // lds_addr
  g0.z = (unsigned)(ga & 0xFFFFFFFFull);                 // global_addr[31:0]
  g0.w = (unsigned)((ga >> 32) & 0x01FFFFFFull)          // global_addr[56:32]
         | (2u << 30);                                   // type=2 ("image")

  i32x8 g1;
  g1[0] = (int)(2u << 16);                               // data_size=2 (4 bytes)
  g1[1] = (int)((n & 0xFFFFu) << 16);                    // tensor_dim0[15:0]
  g1[2] = (int)(((n >> 16) & 0xFFFFu) | (1u << 16));     // tensor_dim0[31:16], tensor_dim1=1
  g1[3] = (int)((n & 0xFFFFu) << 16);                    // tile_dim0 = n
  g1[4] = 1;                                             // tile_dim1=1, tile_dim2=0
  g1[5] = (int)n;                                        // tensor_dim0_stride[31:0]
  g1[6] = 0;                                             // stride hi / dim1_stride lo
  g1[7] = 0;                                             // dim1_stride hi

  i32x4 gz = {0, 0, 0, 0};
#if __clang_major__ >= 23
  i32x8 gz8 = {0, 0, 0, 0, 0, 0, 0, 0};
  __builtin_amdgcn_tensor_load_to_lds(g0, g1, gz, gz, gz8, 0);
#else
  __builtin_amdgcn_tensor_load_to_lds(g0, g1, gz, gz, 0);
#endif
}

__device__ __forceinline__ float iou_fn(float px, float py, float pw, float ph,
                                        float tx, float ty, float tw, float th) {
  // DIV = 448/7 = 64, IMG = 448
  float x1 = px * 64.0f,  y1 = py * 64.0f;
  float w1 = pw * 448.0f, h1 = ph * 448.0f;
  float x2 = tx * 64.0f,  y2 = ty * 64.0f;
  float w2 = tw * 448.0f, h2 = th * 448.0f;
  float xl = fmaxf(x1 - 0.5f * w1, x2 - 0.5f * w2);
  float yt = fmaxf(y1 - 0.5f * h1, y2 - 0.5f * h2);
  float xr = fminf(x1 + 0.5f * w1, x2 + 0.5f * w2);
  float yb = fminf(y1 + 0.5f * h1, y2 + 0.5f * h2);
  float inter = (xr - xl) * (yb - yt);
  float uni   = w1 * h1 + w2 * h2 - inter;
  bool  valid = (xr >= xl) && (yb >= yt);
  return valid ? (inter / uni) : 0.0f;
}

__global__ void __launch_bounds__(256)
yolo_zero_ws(float* ws) {
  if (threadIdx.x < 4) ws[threadIdx.x] = 0.0f;
}

__global__ void __launch_bounds__(256)
yolo_loss_main(const float* __restrict__ pred,
               const float* __restrict__ targ,
               float* __restrict__ ws) {
  __shared__ float sP[2][FLOATS_PER_TILE];
  __shared__ float sT[2][FLOATS_PER_TILE];
  __shared__ float red[8][4];

  const int tid  = threadIdx.x;
  const int wave = tid >> 5;
  const int lane = tid & 31;
  const int tile0 = blockIdx.x * TILES_PER_BLOCK;

  const unsigned ldsP[2] = { (unsigned)(uintptr_t)(void*)&sP[0][0],
                             (unsigned)(uintptr_t)(void*)&sP[1][0] };
  const unsigned ldsT[2] = { (unsigned)(uintptr_t)(void*)&sT[0][0],
                             (unsigned)(uintptr_t)(void*)&sT[1][0] };

  // Prologue: wave 0 kicks off tile 0 into buffer 0 (TENSORcnt = 2).
  if (wave == 0) {
    tdm_load_tile(pred + (size_t)tile0 * FLOATS_PER_TILE, ldsP[0]);
    tdm_load_tile(targ + (size_t)tile0 * FLOATS_PER_TILE, ldsT[0]);
  }

  float accLoc = 0.0f, accObj = 0.0f, accCls = 0.0f, accNp = 0.0f;

  for (int it = 0; it < TILES_PER_BLOCK; ++it) {
    const int buf = it & 1;
    if (wave == 0) {
      if (it + 1 < TILES_PER_BLOCK) {
        // Prefetch next tile into the other buffer, then wait until the two
        // oldest (current tile's) DMAs are done: TENSORcnt 4 -> <= 2.
        const int nb = (it + 1) & 1;
        const size_t goff = (size_t)(tile0 + it + 1) * FLOATS_PER_TILE;
        tdm_load_tile(pred + goff, ldsP[nb]);
        tdm_load_tile(targ + goff, ldsT[nb]);
        __builtin_amdgcn_s_wait_tensorcnt(2);
      } else {
        __builtin_amdgcn_s_wait_tensorcnt(0);
      }
    }
    __syncthreads();  // tile `it` staged in LDS for all 8 waves

    const float* p = &sP[buf][tid * FEAT];
    const float* t = &sT[buf][tid * FEAT];

    const float mask = (t[4] > 0.0f) ? 1.0f : 0.0f;

    const float iou1 = iou_fn(p[0], p[1], p[2], p[3], t[0], t[1], t[2], t[3]);
    const float iou2 = iou_fn(p[5], p[6], p[7], p[8], t[5], t[6], t[7], t[8]);

    const float fx1 = t[0] - floorf(t[0]);
    const float fy1 = t[1] - floorf(t[1]);
    const float fx2 = t[5] - floorf(t[5]);
    const float fy2 = t[6] - floorf(t[6]);

    float dx, dy, dw, dh;
    dx = p[0] - fx1; dy = p[1] - fy1; dw = p[2] - t[2]; dh = p[3] - t[3];
    const float loc1 = 0.5f * (dx * dx + dy * dy) + 0.5f * (dw * dw + dh * dh);
    dx = p[5] - fx2; dy = p[6] - fy2; dw = p[7] - t[7]; dh = p[8] - t[8];
    const float loc2 = 0.5f * (dx * dx + dy * dy) + 0.5f * (dw * dw + dh * dh);

    const float e1 = p[4] - iou1;
    const float e2 = p[9] - iou2;
    const float obj1 = e1 * e1 + 0.5f * p[9] * p[9];
    const float obj2 = e2 * e2 + 0.5f * p[4] * p[4];

    const bool use1 = iou1 > iou2;
    accLoc += mask * (use1 ? loc1 : loc2);
    accObj += mask * (use1 ? obj1 : obj2);
    accNp  += mask;

    const float c0 = p[10] - t[10];
    const float c1 = p[11] - t[11];
    const float c2 = p[12] - t[12];
    accCls += mask * (c0 * c0 + c1 * c1 + c2 * c2);

    __syncthreads();  // all waves done with buf before it is overwritten
  }

  // wave32 tree reduction
  for (int off = 16; off > 0; off >>= 1) {
    accLoc += __shfl_down(accLoc, off, 32);
    accObj += __shfl_down(accObj, off, 32);
    accCls += __shfl_down(accCls, off, 32);
    accNp  += __shfl_down(accNp,  off, 32);
  }
  if (lane == 0) {
    red[wave][0] = accLoc; red[wave][1] = accObj;
    red[wave][2] = accCls; red[wave][3] = accNp;
  }
  __syncthreads();
  if (tid == 0) {
    float L = 0.0f, O = 0.0f, C = 0.0f, N = 0.0f;
    for (int w = 0; w < 8; ++w) {
      L += red[w][0]; O += red[w][1]; C += red[w][2]; N += red[w][3];
    }
    atomicAdd(&ws[0], L);
    atomicAdd(&ws[1], O);
    atomicAdd(&ws[2], C);
    atomicAdd(&ws[3], N);
  }
}

__global__ void __launch_bounds__(32)
yolo_finalize(const float* __restrict__ ws, float* __restrict__ out) {
  if (threadIdx.x == 0) {
    const float loc  = ws[0];
    const float obj  = ws[1];
    const float cls  = ws[2];
    const float npos = ws[3];
    out[0] = 5.0f * loc;             // L_COORD * loc_loss
    out[1] = obj;                    // obj_loss
    out[2] = cls / (npos * 3.0f);    // cls_loss
  }
}

extern "C" void kernel_launch(void* const* d_in, const int* in_sizes, int n_in,
                              void* d_out, int out_size, void* d_ws, size_t ws_size,
                              hipStream_t stream) {
  (void)in_sizes; (void)n_in; (void)out_size; (void)ws_size;
  const float* pred = (const float*)d_in[0];
  const float* targ = (const float*)d_in[1];
  float* out = (float*)d_out;
  float* ws  = (float*)d_ws;   // 4 f32 accumulators: loc, obj, cls, n_pos

  yolo_zero_ws<<<1, 256, 0, stream>>>(ws);
  yolo_loss_main<<<NBLOCKS, 256, 0, stream>>>(pred, targ, ws);
  yolo_finalize<<<1, 32, 0, stream>>>(ws, out);
}